// Popcnt_14731737825611
// MI455X (gfx1250) — compile-verified
//
#include <hip/hip_runtime.h>
#include <cstdint>

#define B_ 256
#define P_ 128
#define FINAL_GROUPS_ 256

typedef __attribute__((ext_vector_type(2))) float v2f;
typedef __attribute__((ext_vector_type(8))) float v8f;

__device__ __forceinline__ float fast_sigmoid(float x) {
  // 1 / (1 + e^-x): v_exp_f32 + v_rcp_f32 (TRANS ops, co-execute with VALU)
  return __builtin_amdgcn_rcpf(1.0f + __expf(-x));
}

// ---------- x [B, IN] -> xT [IN, B] (one-time, makes gathers contiguous) ----------
__global__ __launch_bounds__(256) void transpose_kernel(
    const float* __restrict__ in, float* __restrict__ out, int rows, int cols) {
  __shared__ float tile[32][33];
  int c0 = blockIdx.x * 32, r0 = blockIdx.y * 32;
  int tx = threadIdx.x, ty = threadIdx.y;  // block (32, 8)
#pragma unroll
  for (int i = 0; i < 32; i += 8) {
    int r = r0 + ty + i, c = c0 + tx;
    if (r < rows && c < cols) tile[ty + i][tx] = in[(size_t)r * cols + c];
  }
  __syncthreads();
#pragma unroll
  for (int i = 0; i < 32; i += 8) {
    int c = c0 + ty + i, r = r0 + tx;
    if (c < cols && r < rows) out[(size_t)c * rows + r] = tile[tx][ty + i];
  }
}

// ---------- sigmoid(w) precompute: reused by all 256 batches ----------
__global__ __launch_bounds__(256) void sigmoid_kernel(
    const float* __restrict__ in, float* __restrict__ out, int n) {
  int i = blockIdx.x * 256 + threadIdx.x;
  if (i < n) out[i] = fast_sigmoid(in[i]);
}

// ---------- one popcnt layer, transposed activations [O, B] ----------
// Block = 256 threads (8 wave32) = all batch columns; OT outputs per block.
// sel/sw tiles staged to LDS via CDNA5 async-DMA; gather rows prefetched ahead.
template <int OT, bool NORM, bool FINAL>
__global__ __launch_bounds__(256) void layer_kernel(
    const float* __restrict__ src,   // [Nsrc, 256] transposed activations
    const int* __restrict__ sel,     // [O, 128]
    const float* __restrict__ sw,    // [O, 128] = sigmoid(w)
    const float* __restrict__ bias,  // [O]
    const float* __restrict__ gsc,   // layernorm gamma over source rows (NORM)
    const float* __restrict__ bof,   // layernorm beta (NORM)
    const float* __restrict__ mu,    // [256] per-batch mean (NORM)
    const float* __restrict__ rsig,  // [256] per-batch rsqrt(var+eps) (NORM)
    float* __restrict__ dstT,        // [O, 256] output transposed (!FINAL)
    float* __restrict__ psum,        // [O/OT, 256] partial sums (!FINAL)
    float* __restrict__ psq,         // [O/OT, 256] partial sum-squares (!FINAL)
    float* __restrict__ outF)        // [256, 256] final output (FINAL)
{
  __shared__ __align__(16) int   s_sel[OT * P_];
  __shared__ __align__(16) float s_sw[OT * P_];
  __shared__ float s_bias[OT];

  const int tid = threadIdx.x;
  const int blk = blockIdx.x;
  const int b = tid;  // batch column 0..255

  // ---- CDNA5 async-DMA staging of the tile's indices + weights into LDS ----
  const int* selrow = sel + (size_t)blk * (OT * P_);
  const float* swrow = sw + (size_t)blk * (OT * P_);
  uint32_t lds_sel = (uint32_t)(uintptr_t)&s_sel[0];  // low 32 bits == LDS offset
  uint32_t lds_sw = (uint32_t)(uintptr_t)&s_sw[0];
#pragma unroll
  for (int it = 0; it < (OT * P_) / (256 * 4); ++it) {
    uint32_t byteOff = (uint32_t)(tid + it * 256) * 16u;
    asm volatile("global_load_async_to_lds_b128 %0, %1, %2"
                 :: "v"(lds_sel + byteOff), "v"(byteOff), "s"(selrow)
                 : "memory");
    asm volatile("global_load_async_to_lds_b128 %0, %1, %2"
                 :: "v"(lds_sw + byteOff), "v"(byteOff), "s"(swrow)
                 : "memory");
  }
  if (tid < OT) s_bias[tid] = bias[blk * OT + tid];
  asm volatile("s_wait_asynccnt 0x0" ::: "memory");
  __syncthreads();

  // Per-batch normalization constants: v_norm = (v - mu)*rsig*g + be
  //                                           = fma(fma(v, nsc, nsh), g, be)
  float nsc = 0.f, nsh = 0.f;
  if (NORM) {
    float m0 = mu[b], r0 = rsig[b];
    nsc = r0;
    nsh = -m0 * r0;
  }

  float groupAcc = 0.f, ps = 0.f, pq = 0.f;
#pragma unroll 1
  for (int m = 0; m < OT; ++m) {
    const int* sl = &s_sel[m * P_];
    const float* sv = &s_sw[m * P_];
    float acc = 0.f;
#pragma unroll 4
    for (int p = 0; p < P_; ++p) {
      // Index is wave-uniform (uniform LDS address) -> SGPR addressing
      int j = __builtin_amdgcn_readfirstlane(sl[p]);
      float wv = sv[p];
      if (p + 16 < P_) {  // pull upcoming gather row into WGP$ (global_prefetch_b8)
        int jn = sl[p + 16];
        __builtin_prefetch(src + (size_t)jn * B_ + b, 0, 3);
      }
      float v = src[j * B_ + b];  // contiguous 128B per wave
      if (NORM) v = fmaf(fmaf(v, nsc, nsh), gsc[j], bof[j]);
      acc = fmaf(wv, v, acc);
    }
    float h = fast_sigmoid(acc - s_bias[m]);
    if (FINAL) {
      groupAcc += h;  // OT==16 aligns exactly with one output group
    } else {
      dstT[(size_t)(blk * OT + m) * B_ + b] = h;
      ps += h;
      pq = fmaf(h, h, pq);
    }
  }

  if (FINAL) {
    outF[(size_t)b * FINAL_GROUPS_ + blk] = groupAcc - 8.0f;
  } else {
    psum[(size_t)blk * B_ + b] = ps;
    psq[(size_t)blk * B_ + b] = pq;
  }
}

// ---------- layernorm stats finalize via WMMA column-sums ----------
// psum/psq are [nblk, 256]; we need per-column totals. With A = ones(16x4),
// V_WMMA_F32_16X16X4_F32 computes D[m,n] += sum_k B[k,n]: a 4-row column-sum
// per instruction, accumulated in C. Result read from D row 0 (VGPR0,
// lanes 0..15 = N). Since A is all-ones the sum is invariant to how B's K
// values are split across lane halves; only the N = lane&15 mapping matters.
__global__ __launch_bounds__(32) void stats_wmma_kernel(
    const float* __restrict__ psum, const float* __restrict__ psq, int nblk,
    float invN, float* __restrict__ mu, float* __restrict__ rsig) {
  const int lane = threadIdx.x;         // one full wave32: EXEC all ones
  const int colbase = blockIdx.x * 16;  // 16 batch columns per block
  const int n = lane & 15;
  const int khalf = (lane >> 4) << 1;  // lanes 0-15: K=0,1; lanes 16-31: K=2,3
  const float* ps = psum + colbase + n;
  const float* pq = psq + colbase + n;

#if __has_builtin(__builtin_amdgcn_wmma_f32_16x16x4_f32)
  v2f a;
  a.x = 1.0f;
  a.y = 1.0f;
  v8f cs = {};
  v8f cq = {};
  for (int i = 0; i < nblk; i += 4) {
    v2f bs, bq;
    bs.x = ps[(size_t)(i + khalf) * B_];
    bs.y = ps[(size_t)(i + khalf + 1) * B_];
    bq.x = pq[(size_t)(i + khalf) * B_];
    bq.y = pq[(size_t)(i + khalf + 1) * B_];
    cs = __builtin_amdgcn_wmma_f32_16x16x4_f32(false, a, false, bs, (short)0,
                                               cs, false, false);
    cq = __builtin_amdgcn_wmma_f32_16x16x4_f32(false, a, false, bq, (short)0,
                                               cq, false, false);
  }
  float s = cs[0];  // D[M=0, N=lane] for lanes 0..15
  float q = cq[0];
#else
  float s = 0.f, q = 0.f;
  if (lane < 16) {
    for (int i = 0; i < nblk; ++i) {
      s += ps[(size_t)i * B_];
      q += pq[(size_t)i * B_];
    }
  }
#endif

  if (lane < 16) {
    float m = s * invN;
    float v = fmaf(-m, m, q * invN);
    v = fmaxf(v, 0.0f);
    mu[colbase + n] = m;
    rsig[colbase + n] = __builtin_amdgcn_rsqf(v + 1e-12f);
  }
}

// ---------- workspace layout ----------
static constexpr size_t OFF_XT = 0;                           // 3200*256*4
static constexpr size_t OFF_SW1 = OFF_XT + 3276800;           // 8192*128*4
static constexpr size_t OFF_SW2 = OFF_SW1 + 4194304;          // 8192*128*4
static constexpr size_t OFF_SW3 = OFF_SW2 + 4194304;          // 4096*128*4
static constexpr size_t OFF_H1 = OFF_SW3 + 2097152;           // 8192*256*4
static constexpr size_t OFF_H2 = OFF_H1 + 8388608;            // 8192*256*4
static constexpr size_t OFF_PS1 = OFF_H2 + 8388608;           // 1024*256*4
static constexpr size_t OFF_PQ1 = OFF_PS1 + 1048576;
static constexpr size_t OFF_PS2 = OFF_PQ1 + 1048576;
static constexpr size_t OFF_PQ2 = OFF_PS2 + 1048576;
static constexpr size_t OFF_MU1 = OFF_PQ2 + 1048576;
static constexpr size_t OFF_RS1 = OFF_MU1 + 1024;
static constexpr size_t OFF_MU2 = OFF_RS1 + 1024;
static constexpr size_t OFF_RS2 = OFF_MU2 + 1024;
static constexpr size_t WS_NEEDED = OFF_RS2 + 1024;  // ~33.2 MB

extern "C" void kernel_launch(void* const* d_in, const int* in_sizes, int n_in,
                              void* d_out, int out_size, void* d_ws, size_t ws_size,
                              hipStream_t stream) {
  (void)in_sizes; (void)n_in; (void)out_size;
  if (ws_size < WS_NEEDED) return;

  const float* x = (const float*)d_in[0];
  const int* sel1 = (const int*)d_in[1];
  const float* w1 = (const float*)d_in[2];
  const float* b1 = (const float*)d_in[3];
  const float* g1 = (const float*)d_in[4];
  const float* be1 = (const float*)d_in[5];
  const int* sel2 = (const int*)d_in[6];
  const float* w2 = (const float*)d_in[7];
  const float* b2 = (const float*)d_in[8];
  const float* g2 = (const float*)d_in[9];
  const float* be2 = (const float*)d_in[10];
  const int* sel3 = (const int*)d_in[11];
  const float* w3 = (const float*)d_in[12];
  const float* b3 = (const float*)d_in[13];

  char* ws = (char*)d_ws;
  float* xT  = (float*)(ws + OFF_XT);
  float* sw1 = (float*)(ws + OFF_SW1);
  float* sw2 = (float*)(ws + OFF_SW2);
  float* sw3 = (float*)(ws + OFF_SW3);
  float* h1  = (float*)(ws + OFF_H1);
  float* h2  = (float*)(ws + OFF_H2);
  float* ps1 = (float*)(ws + OFF_PS1);
  float* pq1 = (float*)(ws + OFF_PQ1);
  float* ps2 = (float*)(ws + OFF_PS2);
  float* pq2 = (float*)(ws + OFF_PQ2);
  float* mu1 = (float*)(ws + OFF_MU1);
  float* rs1 = (float*)(ws + OFF_RS1);
  float* mu2 = (float*)(ws + OFF_MU2);
  float* rs2 = (float*)(ws + OFF_RS2);

  // 1) transpose x -> xT [3200, 256]
  transpose_kernel<<<dim3(100, 8), dim3(32, 8), 0, stream>>>(x, xT, 256, 3200);

  // 2) precompute sigmoid(w) per layer (amortized over 256 batches)
  sigmoid_kernel<<<4096, 256, 0, stream>>>(w1, sw1, 8192 * 128);
  sigmoid_kernel<<<4096, 256, 0, stream>>>(w2, sw2, 8192 * 128);
  sigmoid_kernel<<<2048, 256, 0, stream>>>(w3, sw3, 4096 * 128);

  // 3) layer 1: x gather -> h1 [8192, 256] + fused layernorm partial stats
  layer_kernel<8, false, false><<<1024, 256, 0, stream>>>(
      xT, sel1, sw1, b1, nullptr, nullptr, nullptr, nullptr, h1, ps1, pq1, nullptr);
  stats_wmma_kernel<<<16, 32, 0, stream>>>(ps1, pq1, 1024, 1.0f / 8192.0f, mu1, rs1);

  // 4) layer 2: layernorm(h1) fused into gather -> h2 [8192, 256]
  layer_kernel<8, true, false><<<1024, 256, 0, stream>>>(
      h1, sel2, sw2, b2, g1, be1, mu1, rs1, h2, ps2, pq2, nullptr);
  stats_wmma_kernel<<<16, 32, 0, stream>>>(ps2, pq2, 1024, 1.0f / 8192.0f, mu2, rs2);

  // 5) layer 3: layernorm(h2) fused; 16-output tiles == output groups; -8 bias
  layer_kernel<16, true, true><<<256, 256, 0, stream>>>(
      h2, sel3, sw3, b3, g2, be2, mu2, rs2, nullptr, nullptr, nullptr,
      (float*)d_out);
}